// CSRN_76922864272117
// MI455X (gfx1250) — compile-verified
//
#include <hip/hip_runtime.h>
#include <math.h>

// ---------------------------------------------------------------------------
// CDNA5 (gfx1250) implementation of the 4-direction spatial GRU ("CSRN").
// Only 3 directions contribute to the output (the reference discards the left
// pass and zeros ctx_right), so we run down/up/right scans only.
// All GEMMs use V_WMMA_F32_16X16X4_F32 (full fp32 matrix pipe) to preserve
// precision across the 64-step recurrence.
//
// Round-4: per-step grids are smaller than the machine (latency-bound), so
// kernels take __launch_bounds__(32,1) (relax VGPR cap) and explicitly
// double-buffer all WMMA fragments: chunk k+1's loads are issued before
// chunk k's 12 WMMAs, decoupling vmem latency from the matrix pipe.
// ---------------------------------------------------------------------------

typedef __attribute__((ext_vector_type(2))) float v2f;
typedef __attribute__((ext_vector_type(8))) float v8f;

#define WMMA_F32(a, b, c) \
  __builtin_amdgcn_wmma_f32_16x16x4_f32(false, (a), false, (b), (short)0, (c), false, false)

constexpr int kB  = 8;
constexpr int kC  = 256;
constexpr int kH  = 64;
constexpr int kW  = 64;
constexpr int kL  = 64;            // line length per direction
constexpr int kT  = 64;            // steps per scan
constexpr int kBL = kB * kL;       // 512 rows per GRU GEMM
constexpr int kG  = 3 * kC;        // 768 gate columns
constexpr int kHW = kH * kW;       // channel stride in [B,C,H,W]
constexpr int kXsize = kB * kC * kH * kW;  // 8,388,608 floats per ctx buffer

// Workspace layout (float offsets)
constexpr size_t OFF_WIHT = 0;                         // 3 * 256*768 (packed)
constexpr size_t OFF_WHHT = OFF_WIHT + 3ull * kC * kG; // 3 * 256*768 (packed)
constexpr size_t OFF_CWT  = OFF_WHHT + 3ull * kC * kG; // 3 * 3*256*256 (packed)
constexpr size_t OFF_WCT  = OFF_CWT + 3ull * 3 * kC * kC; // 3*256*256 (packed)
constexpr size_t OFF_H    = OFF_WCT + 3ull * kC * kC;  // 3 * 512*256
constexpr size_t OFF_CTX  = OFF_H + 3ull * kBL * kC;   // 3 * kXsize

__device__ __forceinline__ float sigmoidf_(float x) {
  return 1.0f / (1.0f + __expf(-x));
}

// ---------------------------------------------------------------------------
// Weight preparation into fragment-ready packed layouts.
// Packed index for a logical (k, o) in an [K x O] K-major matrix:
//     p(k, o; O) = ((k/4)*O + o)*4 + (k%4)
// so a WMMA lane (half, lq) loads its (ka, ka+1) pair as one b64 from
//     base + ((k0/4)*O + o)*4 + half*2.
//   wihTp/whhTp : [dir] p(k in 256, o in 768)   <- wih/whh [768][256]
//   cwTp        : [dir][kk] p(i in 256, o in 256) <- cw [256][256][3]
//   wcTp        : [cbuf]    p(c in 256, o in 256) <- w_comb [256][1024]
// gridDim = (768, 10), 256 threads. 196608 elements per z-slot.
// ---------------------------------------------------------------------------
__global__ void csrn_prep(const float* __restrict__ wih0, const float* __restrict__ wih1,
                          const float* __restrict__ wih2, const float* __restrict__ whh0,
                          const float* __restrict__ whh1, const float* __restrict__ whh2,
                          const float* __restrict__ cw0, const float* __restrict__ cw1,
                          const float* __restrict__ cw2, const float* __restrict__ wcomb,
                          float* __restrict__ ws) {
  const int idx = blockIdx.x * 256 + threadIdx.x;   // [0, 196608)
  const int z = blockIdx.y;
  if (z < 6) {                                       // wihTp / whhTp
    const int d = (z < 3) ? z : z - 3;
    const float* src;
    if (z < 3) src = (d == 0) ? wih0 : (d == 1) ? wih1 : wih2;
    else       src = (d == 0) ? whh0 : (d == 1) ? whh1 : whh2;
    // idx = ((k/4)*768 + o)*4 + k%4
    const int q  = idx / (kG * 4);
    const int rm = idx % (kG * 4);
    const int o  = rm >> 2;
    const int k  = q * 4 + (rm & 3);
    const size_t off = (z < 3) ? OFF_WIHT : OFF_WHHT;
    ws[off + (size_t)d * kC * kG + idx] = src[o * kC + k];
  } else if (z < 9) {                                // cwTp
    const int d = z - 6;
    const float* src = (d == 0) ? cw0 : (d == 1) ? cw1 : cw2;
    const int kk  = idx >> 16;                       // 3 blocks of 65536
    const int id2 = idx & 65535;                     // ((i/4)*256 + o)*4 + i%4
    const int q   = id2 >> 10;
    const int rm  = id2 & 1023;
    const int o   = rm >> 2;
    const int i   = q * 4 + (rm & 3);
    ws[OFF_CWT + (size_t)d * 3 * kC * kC + idx] = src[(o * kC + i) * 3 + kk];
  } else {                                           // wcTp (3 cbuf blocks)
    const int cbuf = idx >> 16;
    const int id2  = idx & 65535;
    const int q    = id2 >> 10;
    const int rm   = id2 & 1023;
    const int o    = rm >> 2;
    const int c    = cbuf * kC + q * 4 + (rm & 3);
    ws[OFF_WCT + idx] = wcomb[o * (4 * kC) + c];
  }
}

__global__ void csrn_hinit(float* __restrict__ h_all) {
  int i = blockIdx.x * 256 + threadIdx.x;            // 1536 * 256 = 393216
  h_all[i] = 0.0f;
}

// ---------------------------------------------------------------------------
// One GRU step for all 3 directions (gridDim.y = dir).
// Per wave: one 32x16 tile of out[512 x 256]; 12 fp32 WMMA accumulators
// (2 row-tiles x {gi_r, gi_z, gi_n, gh_r, gh_z, gh_n}), K=256 in chunks of 4,
// with explicit fragment double-buffering across chunks.
// ---------------------------------------------------------------------------
struct GruFrag {
  v2f ax0, ax1, ah0, ah1, br, bz, bn, cr, cz, cn;
};

__global__ void __launch_bounds__(32, 1)
csrn_gru_step(const float* __restrict__ x,
              const float* __restrict__ ws_wihT,
              const float* __restrict__ ws_whhT,
              float* __restrict__ h_all,
              float* __restrict__ ctx_all,
              int t) {
  const int dir  = blockIdx.y;
  const int lane = threadIdx.x;
  const int half = lane >> 4;     // selects K pair (A/B), M half (C/D)
  const int lq   = lane & 15;     // M row (A) / N col (B, C/D)
  const int t_eff = (dir == 0) ? t : (kT - 1 - t);

  const int tile    = blockIdx.x;         // 256 tiles = 16 rowPairs x 16 colTiles
  const int rowPair = tile >> 4;
  const int colTile = tile & 15;
  const int row0 = rowPair * 32;          // 32 rows, within one 64-row (b, line)
  const int b    = row0 >> 6;
  const int l0   = row0 & 63;             // 0 or 32
  const int c0   = colTile * 16;

  float* h   = h_all + (size_t)dir * kBL * kC;
  float* ctx = ctx_all + (size_t)dir * kXsize;

  // Uniform, loop-invariant addressing: element (c, l) of this (b, line) slab
  // sits at  base + c*kHW + l*lstride.
  const int    lstride = (dir < 2) ? 1 : kW;
  const size_t base    = (dir < 2) ? ((size_t)(b * kC * kH + t_eff) * kW)
                                   : ((size_t)b * kC * kHW + (size_t)t_eff);

  // Per-lane A-row pointers (two row-tiles); + ka*kHW inside the loop.
  const float* xrow0 = x + base + (size_t)(l0 + lq) * lstride;
  const float* xrow1 = xrow0 + (size_t)16 * lstride;
  const float* hrow0 = h + (size_t)(row0 + lq) * kC;
  const float* hrow1 = hrow0 + (size_t)16 * kC;

  // Packed B pointers: + (k0/4)*kG*4 inside the loop (stride 12 KB).
  const float* wip = ws_wihT + (size_t)dir * kC * kG + (size_t)(c0 + lq) * 4 + half * 2;
  const float* whp = ws_whhT + (size_t)dir * kC * kG + (size_t)(c0 + lq) * 4 + half * 2;

  auto ldf = [&](int k0) {
    GruFrag f;
    const int ka = k0 + half * 2;
    const size_t wq = (size_t)(k0 >> 2) * (kG * 4);
    f.ax0.x = xrow0[(size_t)ka * kHW];
    f.ax0.y = xrow0[(size_t)(ka + 1) * kHW];
    f.ax1.x = xrow1[(size_t)ka * kHW];
    f.ax1.y = xrow1[(size_t)(ka + 1) * kHW];
    f.ah0 = *(const v2f*)&hrow0[ka];
    f.ah1 = *(const v2f*)&hrow1[ka];
    f.br = *(const v2f*)&wip[wq];
    f.bz = *(const v2f*)&wip[wq + 1024];
    f.bn = *(const v2f*)&wip[wq + 2048];
    f.cr = *(const v2f*)&whp[wq];
    f.cz = *(const v2f*)&whp[wq + 1024];
    f.cn = *(const v2f*)&whp[wq + 2048];
    return f;
  };

  v8f air0 = {}, aiz0 = {}, ain0 = {}, ahr0 = {}, ahz0 = {}, ahn0 = {};
  v8f air1 = {}, aiz1 = {}, ain1 = {}, ahr1 = {}, ahz1 = {}, ahn1 = {};

  auto mm = [&](const GruFrag& f) {
    air0 = WMMA_F32(f.ax0, f.br, air0);
    air1 = WMMA_F32(f.ax1, f.br, air1);
    aiz0 = WMMA_F32(f.ax0, f.bz, aiz0);
    aiz1 = WMMA_F32(f.ax1, f.bz, aiz1);
    ain0 = WMMA_F32(f.ax0, f.bn, ain0);
    ain1 = WMMA_F32(f.ax1, f.bn, ain1);
    ahr0 = WMMA_F32(f.ah0, f.cr, ahr0);
    ahr1 = WMMA_F32(f.ah1, f.cr, ahr1);
    ahz0 = WMMA_F32(f.ah0, f.cz, ahz0);
    ahz1 = WMMA_F32(f.ah1, f.cz, ahz1);
    ahn0 = WMMA_F32(f.ah0, f.cn, ahn0);
    ahn1 = WMMA_F32(f.ah1, f.cn, ahn1);
  };

  // Software pipeline: loads of chunk k+4 are issued before WMMAs of chunk k.
  GruFrag f = ldf(0);
#pragma unroll 2
  for (int k0 = 0; k0 < kC - 4; k0 += 4) {
    GruFrag fn = ldf(k0 + 4);
    mm(f);
    f = fn;
  }
  mm(f);

  // Gate math. C/D layout: VGPR j -> M = half*8 + j, N = lq.
  const int c = c0 + lq;
  float* ctxcol = ctx + base + (size_t)c * kHW;   // + l*lstride per element
#pragma unroll
  for (int j = 0; j < 8; ++j) {
    const int m0  = half * 8 + j;
    {
      const int row = row0 + m0;
      const float hv = h[row * kC + c];
      const float r  = sigmoidf_(air0[j] + ahr0[j]);
      const float zg = sigmoidf_(aiz0[j] + ahz0[j]);
      const float nn = tanhf(ain0[j] + r * ahn0[j]);
      ctxcol[(size_t)(l0 + m0) * lstride] = (1.0f - zg) * nn + zg * hv;
    }
    {
      const int row = row0 + 16 + m0;
      const float hv = h[row * kC + c];
      const float r  = sigmoidf_(air1[j] + ahr1[j]);
      const float zg = sigmoidf_(aiz1[j] + ahz1[j]);
      const float nn = tanhf(ain1[j] + r * ahn1[j]);
      ctxcol[(size_t)(l0 + 16 + m0) * lstride] = (1.0f - zg) * nn + zg * hv;
    }
  }
}

// ---------------------------------------------------------------------------
// h_new = tanh(conv1d(ctx, cw, cb)) as 3 shifted K=256 WMMA GEMMs.
// Two row-tiles per wave share the (b64-packed) conv-weight B-fragments.
// Boundary handling: per-tap clamped base pointer + mask multiply.
// Fragments double-buffered across K-chunks.
// ---------------------------------------------------------------------------
struct ConvFrag {
  v2f a0[3], a1[3], bb[3];
};

__global__ void __launch_bounds__(32, 1)
csrn_conv_step(const float* __restrict__ ctx_all,
               const float* __restrict__ ws_cwT,
               const float* __restrict__ cb0,
               const float* __restrict__ cb1,
               const float* __restrict__ cb2,
               float* __restrict__ h_all,
               int t) {
  const int dir  = blockIdx.y;
  const int lane = threadIdx.x;
  const int half = lane >> 4;
  const int lq   = lane & 15;
  const int t_eff = (dir == 0) ? t : (kT - 1 - t);

  const int tile    = blockIdx.x;         // 256 tiles = 16 rowPairs x 16 colTiles
  const int rowPair = tile >> 4;
  const int colTile = tile & 15;
  const int row0 = rowPair * 32;
  const int b    = row0 >> 6;
  const int l0   = row0 & 63;
  const int o0   = colTile * 16;

  const float* ctx = ctx_all + (size_t)dir * kXsize;
  const float* cb  = (dir == 0) ? cb0 : (dir == 1) ? cb1 : cb2;
  float* h = h_all + (size_t)dir * kBL * kC;

  const int    lstride = (dir < 2) ? 1 : kW;
  const size_t base    = (dir < 2) ? ((size_t)(b * kC * kH + t_eff) * kW)
                                   : ((size_t)b * kC * kHW + (size_t)t_eff);

  // Per-tap, per-row-tile A pointers + boundary masks (loop-invariant).
  const float* arow[2][3];
  float msk[2][3];
#pragma unroll
  for (int r = 0; r < 2; ++r) {
#pragma unroll
    for (int kk = 0; kk < 3; ++kk) {
      const int lp = l0 + r * 16 + lq + kk - 1;       // shifted line position
      msk[r][kk] = (lp >= 0 && lp < kL) ? 1.0f : 0.0f;
      const int lc = lp < 0 ? 0 : (lp > kL - 1 ? kL - 1 : lp);
      arow[r][kk] = ctx + base + (size_t)lc * lstride;  // + ka*kHW in the loop
    }
  }

  // Packed conv-weight pointers per tap: + (k0/4)*kC*4 in the loop (4 KB).
  const float* wkp[3];
#pragma unroll
  for (int kk = 0; kk < 3; ++kk)
    wkp[kk] = ws_cwT + ((size_t)dir * 3 + kk) * kC * kC + (size_t)(o0 + lq) * 4 + half * 2;

  auto ldf = [&](int k0) {
    ConvFrag f;
    const int ka = k0 + half * 2;
    const size_t wq = (size_t)(k0 >> 2) * (kC * 4);
#pragma unroll
    for (int kk = 0; kk < 3; ++kk) {
      f.a0[kk].x = arow[0][kk][(size_t)ka * kHW] * msk[0][kk];
      f.a0[kk].y = arow[0][kk][(size_t)(ka + 1) * kHW] * msk[0][kk];
      f.a1[kk].x = arow[1][kk][(size_t)ka * kHW] * msk[1][kk];
      f.a1[kk].y = arow[1][kk][(size_t)(ka + 1) * kHW] * msk[1][kk];
      f.bb[kk] = *(const v2f*)&wkp[kk][wq];
    }
    return f;
  };

  v8f acc0 = {}, acc1 = {};
  auto mm = [&](const ConvFrag& f) {
#pragma unroll
    for (int kk = 0; kk < 3; ++kk) {
      acc0 = WMMA_F32(f.a0[kk], f.bb[kk], acc0);
      acc1 = WMMA_F32(f.a1[kk], f.bb[kk], acc1);
    }
  };

  ConvFrag f = ldf(0);
#pragma unroll 2
  for (int k0 = 0; k0 < kC - 4; k0 += 4) {
    ConvFrag fn = ldf(k0 + 4);
    mm(f);
    f = fn;
  }
  mm(f);

  const int o = o0 + lq;
  const float bias = cb[o];
#pragma unroll
  for (int j = 0; j < 8; ++j) {
    const int m = half * 8 + j;
    h[(row0 + m) * kC + o]      = tanhf(acc0[j] + bias);
    h[(row0 + 16 + m) * kC + o] = tanhf(acc1[j] + bias);
  }
}

// ---------------------------------------------------------------------------
// Final combine: out[b,o,h,w] = b_comb[o] + sum over 3 ctx buffers of
// ctx[b,c,h,w] * w_comb[o, buf*256 + c]  (ctx_right contributes zero).
// One wave per 32x16 output tile (two row-tiles share b64 B), K = 768,
// fragments double-buffered within each 256-wide ctx block.
// ---------------------------------------------------------------------------
struct CombFrag {
  v2f a0, a1, bb;
};

__global__ void __launch_bounds__(32, 1)
csrn_combine(const float* __restrict__ ctx_all,
             const float* __restrict__ wcT,
             const float* __restrict__ b_comb,
             float* __restrict__ out) {
  const int lane = threadIdx.x;
  const int half = lane >> 4;
  const int lq   = lane & 15;

  const int tile    = blockIdx.x;          // 1024 rowPairs x 16 colTiles = 16384
  const int rowPair = tile >> 4;
  const int colTile = tile & 15;
  const int row0 = rowPair * 32;           // rows = (b*H + h)*W + w
  const int b  = row0 >> 12;
  const int hh = (row0 >> 6) & 63;
  const int w0 = row0 & 63;                // 0 or 32
  const int o0 = colTile * 16;

  const size_t base = (size_t)(b * kC * kH + hh) * kW + w0 + lq;

  v8f acc0 = {}, acc1 = {};
#pragma unroll
  for (int cbuf = 0; cbuf < 3; ++cbuf) {
    const float* arow0 = ctx_all + (size_t)cbuf * kXsize + base;
    const float* arow1 = arow0 + 16;       // +16 along w
    const float* wbase = wcT + (size_t)cbuf * kC * kC + (size_t)(o0 + lq) * 4 + half * 2;

    auto ldf = [&](int k0) {
      CombFrag f;
      const int ka = k0 + half * 2;
      f.a0.x = arow0[(size_t)ka * kHW];
      f.a0.y = arow0[(size_t)(ka + 1) * kHW];
      f.a1.x = arow1[(size_t)ka * kHW];
      f.a1.y = arow1[(size_t)(ka + 1) * kHW];
      f.bb = *(const v2f*)&wbase[(size_t)(k0 >> 2) * (kC * 4)];
      return f;
    };
    auto mm = [&](const CombFrag& f) {
      acc0 = WMMA_F32(f.a0, f.bb, acc0);
      acc1 = WMMA_F32(f.a1, f.bb, acc1);
    };

    CombFrag f = ldf(0);
#pragma unroll 4
    for (int k0 = 0; k0 < kC - 4; k0 += 4) {
      CombFrag fn = ldf(k0 + 4);
      mm(f);
      f = fn;
    }
    mm(f);
  }

  const int o = o0 + lq;
  const float bias = b_comb[o];
  float* orow = out + (size_t)((b * kC + o) * kH + hh) * kW;
#pragma unroll
  for (int j = 0; j < 8; ++j) {
    const int m = half * 8 + j;
    orow[w0 + m]      = acc0[j] + bias;
    orow[w0 + 16 + m] = acc1[j] + bias;
  }
}

// ---------------------------------------------------------------------------
extern "C" void kernel_launch(void* const* d_in, const int* in_sizes, int n_in,
                              void* d_out, int out_size, void* d_ws, size_t ws_size,
                              hipStream_t stream) {
  (void)in_sizes; (void)n_in; (void)out_size; (void)ws_size;
  const float* x = (const float*)d_in[0];
  // direction mapping: 0=down(d), 1=up(u), 2=right(r). Left pass is unused.
  const float* wih0 = (const float*)d_in[1];
  const float* whh0 = (const float*)d_in[2];
  const float* cw0  = (const float*)d_in[3];
  const float* cb0  = (const float*)d_in[4];
  const float* wih1 = (const float*)d_in[5];
  const float* whh1 = (const float*)d_in[6];
  const float* cw1  = (const float*)d_in[7];
  const float* cb1  = (const float*)d_in[8];
  const float* wih2 = (const float*)d_in[13];
  const float* whh2 = (const float*)d_in[14];
  const float* cw2  = (const float*)d_in[15];
  const float* cb2  = (const float*)d_in[16];
  const float* wcomb = (const float*)d_in[17];
  const float* bcomb = (const float*)d_in[18];

  float* ws = (float*)d_ws;

  // Weight packing + hidden-state zeroing (every call; deterministic).
  csrn_prep<<<dim3(768, 10), 256, 0, stream>>>(wih0, wih1, wih2, whh0, whh1, whh2,
                                               cw0, cw1, cw2, wcomb, ws);
  csrn_hinit<<<1536, 256, 0, stream>>>(ws + OFF_H);

  for (int t = 0; t < kT; ++t) {
    csrn_gru_step<<<dim3(256, 3), 32, 0, stream>>>(
        x, ws + OFF_WIHT, ws + OFF_WHHT, ws + OFF_H, ws + OFF_CTX, t);
    if (t < kT - 1) {
      csrn_conv_step<<<dim3(256, 3), 32, 0, stream>>>(
          ws + OFF_CTX, ws + OFF_CWT, cb0, cb1, cb2, ws + OFF_H, t);
    }
  }

  csrn_combine<<<16384, 32, 0, stream>>>(ws + OFF_CTX, ws + OFF_WCT, bcomb,
                                         (float*)d_out);
}